// NonMaximaSuppression2d_22840636080729
// MI455X (gfx1250) — compile-verified
//
#include <hip/hip_runtime.h>
#include <stdint.h>

// NMS-2D, 3x3 neighborhood excluding center, replicate padding.
// Input:  x (8,4,2048,2048) fp32.  Output: x * (x > max_3x3_noncenter).
// Memory-bound: ~1 GiB compulsory traffic -> ~46us roofline @ 23.3 TB/s.
// Strategy: CDNA5 async global->LDS staging (global_load_async_to_lds_b128,
// ASYNCcnt), LDS-halo tiles, b128 LDS reads, b128 coalesced stores.

#define IMG_H 2048
#define IMG_W 2048
#define TILE_H 32
#define TILE_W 128
#define LDS_W  136              // 4 halo + 128 + 4 halo (keeps every b128 aligned)
#define LDS_H  34               // 1 halo + 32 + 1 halo
#define NTRANS (LDS_H * (LDS_W / 4))   // 34*34 = 1156 b128 transfers per tile
#define NCHUNK (TILE_H * (TILE_W / 4)) // 1024 float4 outputs per tile

// Address-space-typed int4 pointers for the async-copy builtin.
typedef int v4i __attribute__((ext_vector_type(4)));
typedef __attribute__((address_space(1))) v4i as1_v4i;   // global
typedef __attribute__((address_space(3))) v4i as3_v4i;   // LDS

__global__ __launch_bounds__(256) void nms2d_kernel(const float* __restrict__ x,
                                                    float* __restrict__ out) {
    __shared__ float tile[LDS_H * LDS_W];   // 18,496 B

    const int t  = threadIdx.x;
    const int x0 = blockIdx.x * TILE_W;
    const int y0 = blockIdx.y * TILE_H;
    const size_t plane_off = (size_t)blockIdx.z * (size_t)(IMG_H * IMG_W);
    const float* __restrict__ plane  = x   + plane_off;
    float* __restrict__       oplane = out + plane_off;

    // ---------------- stage: async global -> LDS (b128 per lane) ----------------
    for (int i = t; i < NTRANS; i += 256) {
        const int lr  = i / (LDS_W / 4);        // LDS row   0..33
        const int lc4 = i % (LDS_W / 4);        // float4 col 0..33
        int gy = y0 - 1 + lr;                   // row clamp == replicate padding
        gy = gy < 0 ? 0 : (gy > IMG_H - 1 ? IMG_H - 1 : gy);
        int gx = x0 - 4 + lc4 * 4;              // col clamp: slots fixed in compute
        gx = gx < 0 ? 0 : (gx > IMG_W - 4 ? IMG_W - 4 : gx);
        const float* gp = plane + (size_t)gy * IMG_W + gx;
        // Low 32 bits of a flat LDS pointer are the LDS byte offset (aperture rule).
        const unsigned lds_addr = (unsigned)(uintptr_t)(&tile[lr * LDS_W + lc4 * 4]);
#if __has_builtin(__builtin_amdgcn_global_load_async_to_lds_b128)
        __builtin_amdgcn_global_load_async_to_lds_b128(
            (as1_v4i*)(uintptr_t)gp,
            (as3_v4i*)lds_addr,
            0, 0);
#else
        asm volatile("global_load_async_to_lds_b128 %0, %1, off"
                     :: "v"(lds_addr), "v"(gp)
                     : "memory");
#endif
    }
#if __has_builtin(__builtin_amdgcn_s_wait_asynccnt)
    __builtin_amdgcn_s_wait_asynccnt(0);
#else
    asm volatile("s_wait_asynccnt 0x0" ::: "memory");
#endif
    __syncthreads();   // publish all waves' async LDS writes to the workgroup

    // ---------------- compute: 4 outputs (float4) per thread-iteration ----------
    for (int c = t; c < NCHUNK; c += 256) {
        const int row  = c / (TILE_W / 4);      // 0..31
        const int col4 = c % (TILE_W / 4);      // 0..31
        const int base = 4 + col4 * 4;          // LDS column of first output
        const int gxb  = x0 + col4 * 4;         // global x of first output
        const int gy   = y0 + row;

        const float* __restrict__ rA = &tile[(row + 0) * LDS_W]; // y-1
        const float* __restrict__ rC = &tile[(row + 1) * LDS_W]; // y   (center row)
        const float* __restrict__ rB = &tile[(row + 2) * LDS_W]; // y+1

        const float4 a  = *(const float4*)&rA[base];
        const float4 cc = *(const float4*)&rC[base];
        const float4 b  = *(const float4*)&rB[base];
        float al = rA[base - 1], ar = rA[base + 4];
        float cl = rC[base - 1], cr = rC[base + 4];
        float bl = rB[base - 1], br = rB[base + 4];

        // Replicate-edge redirect: only j=0 can sit at x==0, only j=3 at x==W-1.
        const bool eL = (gxb == 0);
        const bool eR = (gxb == IMG_W - 4);
        al = eL ? a.x  : al;  cl = eL ? cc.x : cl;  bl = eL ? b.x  : bl;
        ar = eR ? a.w  : ar;  cr = eR ? cc.w : cr;  br = eR ? b.w  : br;

        float4 o;
        {   // j = 0
            float m = fmaxf(fmaxf(al,  a.x), a.y);
            m = fmaxf(m, fmaxf(fmaxf(bl, b.x), b.y));
            m = fmaxf(m, fmaxf(cl, cc.y));
            o.x = (cc.x > m) ? cc.x : 0.0f;
        }
        {   // j = 1
            float m = fmaxf(fmaxf(a.x, a.y), a.z);
            m = fmaxf(m, fmaxf(fmaxf(b.x, b.y), b.z));
            m = fmaxf(m, fmaxf(cc.x, cc.z));
            o.y = (cc.y > m) ? cc.y : 0.0f;
        }
        {   // j = 2
            float m = fmaxf(fmaxf(a.y, a.z), a.w);
            m = fmaxf(m, fmaxf(fmaxf(b.y, b.z), b.w));
            m = fmaxf(m, fmaxf(cc.y, cc.w));
            o.z = (cc.z > m) ? cc.z : 0.0f;
        }
        {   // j = 3
            float m = fmaxf(fmaxf(a.z, a.w), ar);
            m = fmaxf(m, fmaxf(fmaxf(b.z, b.w), br));
            m = fmaxf(m, fmaxf(cc.z, cr));
            o.w = (cc.w > m) ? cc.w : 0.0f;
        }
        *(float4*)(oplane + (size_t)gy * IMG_W + gxb) = o;
    }
}

extern "C" void kernel_launch(void* const* d_in, const int* in_sizes, int n_in,
                              void* d_out, int out_size, void* d_ws, size_t ws_size,
                              hipStream_t stream) {
    (void)in_sizes; (void)n_in; (void)d_ws; (void)ws_size; (void)out_size;
    const float* xin = (const float*)d_in[0];
    float* out = (float*)d_out;
    // planes = 8*4 = 32; x-tiles = 2048/128 = 16; y-tiles = 2048/32 = 64
    dim3 grid(IMG_W / TILE_W, IMG_H / TILE_H, 8 * 4);
    nms2d_kernel<<<grid, 256, 0, stream>>>(xin, out);
}